// RipsLayer_35751307772508
// MI455X (gfx1250) — compile-verified
//
#include <hip/hip_runtime.h>
#include <hip/hip_bf16.h>

#define NPTS 4096
#define DFEAT 64
#define TILES_PER_DIM (NPTS / 16)      // 256
#define PRIM_THREADS 512

typedef __attribute__((ext_vector_type(2))) float v2f;
typedef __attribute__((ext_vector_type(8))) float v8f;

// ---------------------------------------------------------------------------
// Kernel 1: squared norms of each row of x.  sq[i] = sum_k x[i][k]^2
// ---------------------------------------------------------------------------
__global__ void sqnorm_kernel(const float* __restrict__ x, float* __restrict__ sq) {
    int i = blockIdx.x * blockDim.x + threadIdx.x;   // 4096 threads total
    const float4* row = (const float4*)(x + (size_t)i * DFEAT);
    float s = 0.0f;
#pragma unroll
    for (int k = 0; k < DFEAT / 4; ++k) {
        float4 v = row[k];
        s += v.x * v.x + v.y * v.y + v.z * v.z + v.w * v.w;
    }
    sq[i] = s;
}

// ---------------------------------------------------------------------------
// Kernel 2: distance matrix via fp32 WMMA (V_WMMA_F32_16X16X4_F32).
// One wave computes one 16x16 tile of Gram = X * X^T, K = 64 = 16 x (K=4).
//
// A (16x4 fp32) per-lane layout (ISA 7.12.2): lanes 0-15 hold M=lane,
// VGPR0=K0,VGPR1=K1; lanes 16-31 hold M=lane-16, VGPR0=K2,VGPR1=K3.
// B (4x16) mirrors this with N in place of M.  Both A and B rows come from
// rows of X (B is X^T's column block == X row block), so the load pattern
// is identical modulo the tile base.
//
// C/D (16x16 f32, 8 VGPRs): acc[r] in lane L is element (M = r + 8*(L>=16),
// N = L&15).
// ---------------------------------------------------------------------------
__global__ void dist_wmma_kernel(const float* __restrict__ x,
                                 const float* __restrict__ sq,
                                 float* __restrict__ D) {
    int gthread = blockIdx.x * blockDim.x + threadIdx.x;
    int wave = gthread >> 5;
    int lane = threadIdx.x & 31;

    int tr = wave / TILES_PER_DIM;      // tile row
    int tc = wave % TILES_PER_DIM;      // tile col
    int rbase = tr * 16;
    int cbase = tc * 16;

    int m    = lane & 15;               // M (for A) / N (for B) within tile
    int koff = (lane & 16) ? 2 : 0;     // lanes 16-31 carry K=2,3 of each slab

    const float* arow = x + (size_t)(rbase + m) * DFEAT + koff;
    const float* brow = x + (size_t)(cbase + m) * DFEAT + koff;

    v8f acc = {};
#pragma unroll
    for (int kk = 0; kk < DFEAT / 4; ++kk) {
        v2f a = *(const v2f*)(arow + 4 * kk);   // K = 4kk+koff, 4kk+koff+1
        v2f b = *(const v2f*)(brow + 4 * kk);
        acc = __builtin_amdgcn_wmma_f32_16x16x4_f32(
            /*neg_a=*/false, a, /*neg_b=*/false, b,
            /*c_mod=*/(short)0, acc, /*reuse_a=*/false, /*reuse_b=*/false);
    }

    // Convert Gram tile -> distance tile and store.
    int n   = cbase + m;                // global column (N = lane&15)
    float sn = sq[n];
    int mhi = (lane & 16) ? 8 : 0;
#pragma unroll
    for (int r = 0; r < 8; ++r) {
        int gm = rbase + r + mhi;       // global row
        float d2 = sq[gm] + sn - 2.0f * acc[r];
        float d;
        if (gm == n) {
            d = __builtin_inff();       // diagonal masked to +inf
        } else {
            d2 = fmaxf(d2, 1e-12f);
            d = sqrtf(d2);
        }
        D[(size_t)gm * NPTS + n] = d;
    }
}

// ---------------------------------------------------------------------------
// Kernel 3: Prim's MST (dim-0 persistence deaths), single workgroup.
// mind / intree live in LDS; argmin uses wave32 shuffle reduction with
// lowest-index tie-breaking to match jnp.argmin.
// ---------------------------------------------------------------------------
__global__ __launch_bounds__(PRIM_THREADS)
void prim_kernel(const float* __restrict__ D, float* __restrict__ out) {
    __shared__ float mind[NPTS];
    __shared__ int   intree[NPTS];
    __shared__ float wv[PRIM_THREADS / 32];
    __shared__ int   wi[PRIM_THREADS / 32];
    __shared__ int   sj;

    const int tid  = threadIdx.x;
    const int lane = tid & 31;
    const int wid  = tid >> 5;
    const float INF = __builtin_inff();

    // init: mind = D[0][:], tree = {0}
    for (int i = tid; i < NPTS; i += PRIM_THREADS) {
        mind[i]   = D[i];
        intree[i] = 0;
    }
    if (tid == 0) intree[0] = 1;
    __syncthreads();

    for (int t = 0; t < NPTS - 1; ++t) {
        // --- local argmin over masked mind ---
        float bv = INF;
        int   bi = 0x7fffffff;
        for (int i = tid; i < NPTS; i += PRIM_THREADS) {
            float v = intree[i] ? INF : mind[i];
            if (v < bv || (v == bv && i < bi)) { bv = v; bi = i; }
        }
        // --- wave32 shuffle reduction (min, lowest index on tie) ---
#pragma unroll
        for (int off = 16; off > 0; off >>= 1) {
            float ov = __shfl_xor(bv, off, 32);
            int   oi = __shfl_xor(bi, off, 32);
            if (ov < bv || (ov == bv && oi < bi)) { bv = ov; bi = oi; }
        }
        if (lane == 0) { wv[wid] = bv; wi[wid] = bi; }
        __syncthreads();
        // --- cross-wave reduction + commit by thread 0 ---
        if (tid == 0) {
            float v = wv[0];
            int   j = wi[0];
#pragma unroll
            for (int w = 1; w < PRIM_THREADS / 32; ++w) {
                if (wv[w] < v || (wv[w] == v && wi[w] < j)) { v = wv[w]; j = wi[w]; }
            }
            sj = j;
            intree[j] = 1;
            out[2 * t + 0] = 0.0f;   // birth
            out[2 * t + 1] = v;      // death
        }
        __syncthreads();
        // --- relax: mind = min(mind, D[j][:]) ---
        const float* row = D + (size_t)sj * NPTS;
        for (int i = tid; i < NPTS; i += PRIM_THREADS)
            mind[i] = fminf(mind[i], row[i]);
        __syncthreads();
    }
}

// ---------------------------------------------------------------------------
extern "C" void kernel_launch(void* const* d_in, const int* in_sizes, int n_in,
                              void* d_out, int out_size, void* d_ws, size_t ws_size,
                              hipStream_t stream) {
    (void)in_sizes; (void)n_in; (void)out_size; (void)ws_size;

    const float* x = (const float*)d_in[0];
    float* out = (float*)d_out;

    // Workspace layout: [ D : 4096*4096 f32 (64 MB, L2-resident) ][ sq : 4096 f32 ]
    float* D  = (float*)d_ws;
    float* sq = D + (size_t)NPTS * NPTS;

    // 1) squared norms
    sqnorm_kernel<<<NPTS / 256, 256, 0, stream>>>(x, sq);

    // 2) distance matrix via WMMA: 256x256 tiles, one wave per tile,
    //    8 waves (256 threads) per block.
    int total_waves  = TILES_PER_DIM * TILES_PER_DIM;   // 65536
    int waves_per_blk = 8;
    dist_wmma_kernel<<<total_waves / waves_per_blk, waves_per_blk * 32, 0, stream>>>(x, sq, D);

    // 3) serial Prim phase, one workgroup, LDS-resident state
    prim_kernel<<<1, PRIM_THREADS, 0, stream>>>(D, out);
}